// Sparsemax_70617852281362
// MI455X (gfx1250) — compile-verified
//
#include <hip/hip_runtime.h>
#include <hip/hip_bf16.h>
#include <stdint.h>

// Sparsemax over rows of [4096, 32000] fp32.
// One workgroup (1024 threads = 32 waves) per row. Row cached in LDS (128 KB)
// via CDNA5 async global->LDS loads (global_load_async_to_lds_b128 + ASYNCcnt);
// tau found by wave32 bisection over the tiny candidate set {z - max > -1}
// (only those can be in the support since tau >= z_max - 1).
// HBM traffic = 1 read + 1 write = roofline-optimal (~45 us at 23.3 TB/s).
// Per-row critical path kept short: only 5 workgroup barriers per row; all
// reductions are wave32 shuffle butterflies (barrier-free).

#define ROWS      4096
#define DCOLS     32000
#define DCOLS4    (DCOLS / 4)      // 8000 float4 per row
#define NTHREADS  1024             // 32 waves
#define NWAVES    (NTHREADS / 32)
#define NCAND     2048             // ~35 expected candidates; huge safety margin
#define BISECT_IT 40               // interval 1.0 -> 2^-40, below f32 eps

#if defined(__has_builtin)
#  if __has_builtin(__builtin_amdgcn_global_load_async_to_lds_b128) && \
      __has_builtin(__builtin_amdgcn_s_wait_asynccnt)
#    define HAVE_ASYNC_LDS 1
#  endif
#endif
#ifndef HAVE_ASYNC_LDS
#  define HAVE_ASYNC_LDS 0
#endif

// Native clang vector types (what the async builtins / nontemporal builtins want)
typedef int   v4i __attribute__((vector_size(16)));
typedef float v4f __attribute__((vector_size(16)));
typedef __attribute__((address_space(1))) v4i gbl_v4i;   // global (AS1)
typedef __attribute__((address_space(3))) v4i lds_v4i;   // LDS (AS3)

__device__ __forceinline__ float wave_max(float m) {
    #pragma unroll
    for (int off = 16; off > 0; off >>= 1)
        m = fmaxf(m, __shfl_xor(m, off, 32));
    return m;
}

__global__ __launch_bounds__(NTHREADS)
void Sparsemax_70617852281362_kernel(const float* __restrict__ logits,
                                     float* __restrict__ out) {
    extern __shared__ __align__(16) float s_row[];   // DCOLS floats (dynamic, 128000 B)
    __shared__ float s_wred[NWAVES];                 // per-wave partial maxima
    __shared__ float s_cand[NCAND];
    __shared__ int   s_count;
    __shared__ float s_max;
    __shared__ float s_tau;

    const int tid  = threadIdx.x;
    const int lane = tid & 31;
    const int wave = tid >> 5;
    const int row  = blockIdx.x;

    const float4* __restrict__ src4 = (const float4*)(logits + (size_t)row * DCOLS);
    float4* __restrict__       dst4 = (float4*)(out + (size_t)row * DCOLS);
    float4* s_row4 = (float4*)s_row;

    if (tid == 0) s_count = 0;

    // ---------------- Stage 1: stream row into LDS (async DMA path) ----------
#if HAVE_ASYNC_LDS
    for (int i = tid; i < DCOLS4; i += NTHREADS) {
        __builtin_amdgcn_global_load_async_to_lds_b128(
            (gbl_v4i*)(src4 + i),             // global source (AS1)
            (lds_v4i*)(s_row4 + i),           // LDS destination (AS3)
            /*imm offset*/ 0, /*cpol*/ 0);
    }
    __builtin_amdgcn_s_wait_asynccnt(0);
#else
    for (int i = tid; i < DCOLS4; i += NTHREADS) {
        s_row4[i] = src4[i];
    }
#endif
    __syncthreads();                                           // barrier #1

    // ---------------- Stage 2: row max (shuffle butterflies, 2 barriers) -----
    float m = -3.402823466e38f;
    for (int i = tid; i < DCOLS4; i += NTHREADS) {
        float4 v = s_row4[i];
        m = fmaxf(m, fmaxf(fmaxf(v.x, v.y), fmaxf(v.z, v.w)));
    }
    m = wave_max(m);
    if (lane == 0) s_wred[wave] = m;
    __syncthreads();                                           // barrier #2
    if (tid < 32) {
        float wm = wave_max(s_wred[tid]);   // NWAVES == 32 exactly
        if (tid == 0) s_max = wm;
    }
    __syncthreads();                                           // barrier #3
    const float rowmax = s_max;

    // ---------------- Stage 3: gather support candidates (z - max > -1) ------
    for (int i = tid; i < DCOLS4; i += NTHREADS) {
        float4 v = s_row4[i];
        float c;
        c = v.x - rowmax; if (c > -1.0f) { int k = atomicAdd(&s_count, 1); if (k < NCAND) s_cand[k] = c; }
        c = v.y - rowmax; if (c > -1.0f) { int k = atomicAdd(&s_count, 1); if (k < NCAND) s_cand[k] = c; }
        c = v.z - rowmax; if (c > -1.0f) { int k = atomicAdd(&s_count, 1); if (k < NCAND) s_cand[k] = c; }
        c = v.w - rowmax; if (c > -1.0f) { int k = atomicAdd(&s_count, 1); if (k < NCAND) s_cand[k] = c; }
    }
    __syncthreads();                                           // barrier #4

    // ---------------- Stage 4: wave0 bisects tau (no barriers, LDS in loop only if K > 128)
    // f(tau) = sum max(c - tau, 0) is strictly decreasing on [-1, 0] with
    // f(-1) >= 1 > 0 = f(0). Non-candidates contribute 0 for tau in [-1,0].
    // Sentinel -2.0f also contributes 0 over the whole interval.
    if (tid < 32) {
        int K = s_count; if (K > NCAND) K = NCAND;
        const float c0 = (tid       < K) ? s_cand[tid      ] : -2.0f;
        const float c1 = (tid + 32  < K) ? s_cand[tid + 32 ] : -2.0f;
        const float c2 = (tid + 64  < K) ? s_cand[tid + 64 ] : -2.0f;
        const float c3 = (tid + 96  < K) ? s_cand[tid + 96 ] : -2.0f;
        const bool  big = (K > 128);

        float lo = -1.0f, hi = 0.0f;
        for (int it = 0; it < BISECT_IT; ++it) {
            float mid = 0.5f * (lo + hi);
            float p = fmaxf(c0 - mid, 0.0f) + fmaxf(c1 - mid, 0.0f)
                    + fmaxf(c2 - mid, 0.0f) + fmaxf(c3 - mid, 0.0f);
            if (big) {
                for (int j = tid + 128; j < K; j += 32)
                    p += fmaxf(s_cand[j] - mid, 0.0f);
            }
            // full butterfly: every lane ends with the identical total
            #pragma unroll
            for (int off = 16; off > 0; off >>= 1)
                p += __shfl_xor(p, off, 32);
            if (p >= 1.0f) lo = mid; else hi = mid;
        }
        if (tid == 0) s_tau = 0.5f * (lo + hi);
    }
    __syncthreads();                                           // barrier #5
    const float tau = s_tau;
    const float shift = rowmax + tau;

    // ---------------- Stage 5: write output (streaming, nontemporal) ---------
    for (int i = tid; i < DCOLS4; i += NTHREADS) {
        float4 v = s_row4[i];
        v4f o;
        o[0] = fmaxf(v.x - shift, 0.0f);
        o[1] = fmaxf(v.y - shift, 0.0f);
        o[2] = fmaxf(v.z - shift, 0.0f);
        o[3] = fmaxf(v.w - shift, 0.0f);
        __builtin_nontemporal_store(o, (v4f*)(dst4 + i));
    }
}

extern "C" void kernel_launch(void* const* d_in, const int* in_sizes, int n_in,
                              void* d_out, int out_size, void* d_ws, size_t ws_size,
                              hipStream_t stream) {
    (void)in_sizes; (void)n_in; (void)d_ws; (void)ws_size; (void)out_size;
    const float* logits = (const float*)d_in[0];
    float* out = (float*)d_out;
    const size_t dyn_lds = (size_t)DCOLS * sizeof(float);   // 128000 B row cache
    Sparsemax_70617852281362_kernel<<<ROWS, NTHREADS, dyn_lds, stream>>>(logits, out);
}